// OhemCrossEntropy2d_4587025072406
// MI455X (gfx1250) — compile-verified
//
#include <hip/hip_runtime.h>
#include <math.h>

// Problem constants (from reference)
#define NB  8
#define CC  19
#define HH  512
#define WWI 1024
#define HW  (HH * WWI)        // 524288 = 2^19
#define OH  64
#define OW  128
#define DSN (NB * OH * OW)    // 65536 downsampled pixels
#define MIN_KEPT_DS 1562      // 100000 // 64
#define THR07 0.7f

typedef __attribute__((ext_vector_type(2))) float v2f;
typedef __attribute__((ext_vector_type(8))) float v8f;

// ---------------------------------------------------------------------------
// Wave32 sum reduction. On gfx1250 we route it through the matrix pipe:
// D = A(16x4) * ones(4x16) + C  => every D[i][j] row-sum; summing the 8
// accumulator VGPRs per lane gives the half-wave total, one shfl_xor(16)
// merges the halves. Deterministic, frees the VALU, emits v_wmma_*.
// ---------------------------------------------------------------------------
__device__ __forceinline__ float wave_sum32(float v) {
#if defined(__gfx1250__) && __has_builtin(__builtin_amdgcn_wmma_f32_16x16x4_f32)
    v2f a;    a[0] = v;    a[1] = 0.0f;
    v2f ones; ones[0] = 1.0f; ones[1] = 1.0f;
    v8f c = {0.f, 0.f, 0.f, 0.f, 0.f, 0.f, 0.f, 0.f};
    c = __builtin_amdgcn_wmma_f32_16x16x4_f32(false, a, false, ones,
                                              (short)0, c, false, false);
    float s = c[0] + c[1] + c[2] + c[3] + c[4] + c[5] + c[6] + c[7];
    return s + __shfl_xor(s, 16, 32);
#else
    for (int off = 16; off > 0; off >>= 1) v += __shfl_xor(v, off, 32);
    return v;
#endif
}

// softmax prob of class cs at one (n,h,w): exp(x_cs - max) / sum(exp(x - max))
__device__ __forceinline__ float prob_cls(const float* __restrict__ pred,
                                          int nbase, int off, int cs) {
    float xs[CC];
    float m = -3.402823466e38f;
#pragma unroll
    for (int c = 0; c < CC; ++c) {
        xs[c] = pred[nbase + c * HW + off];
        m = fmaxf(m, xs[c]);
    }
    float s = 0.f, xt = xs[0];
#pragma unroll
    for (int c = 0; c < CC; ++c) {
        s += expf(xs[c] - m);
        if (c == cs) xt = xs[c];
    }
    return expf(xt - m) / s;
}

// ---------------------------------------------------------------------------
// Kernel 1: downsampled true-class probability (bilinear, H then W, matching
// scipy zoom order=1 composition) + nearest target. One thread per ds pixel.
// ---------------------------------------------------------------------------
__global__ void ohem_ds_pred(const float* __restrict__ pred,
                             const int* __restrict__ tgt,
                             float* __restrict__ P) {
    int t = blockIdx.x * 256 + threadIdx.x;        // 0..65535
    int j = t & (OW - 1);
    int i = (t >> 7) & (OH - 1);
    int n = t >> 13;

    const float SH = (float)(511.0 / 63.0);        // (in-1)/(out-1), f32 like jnp
    const float SW = (float)(1023.0 / 127.0);
    float ch = (float)i * SH;
    float cw = (float)j * SW;

    int   h0 = (int)floorf(ch); int h1 = min(h0 + 1, HH - 1);
    float wh = ch - (float)h0;
    int   w0 = (int)floorf(cw); int w1 = min(w0 + 1, WWI - 1);
    float ww = cw - (float)w0;

    int ih = min((int)rintf(ch), HH - 1);          // rintf = round-half-even (jnp.round)
    int iw = min((int)rintf(cw), WWI - 1);
    int tg = tgt[(n * HH + ih) * WWI + iw];
    int cs = min(max(tg, 0), CC - 1);

    int nbase = n * (CC * HW);
    float p00 = prob_cls(pred, nbase, h0 * WWI + w0, cs);
    float p10 = prob_cls(pred, nbase, h1 * WWI + w0, cs);
    float p01 = prob_cls(pred, nbase, h0 * WWI + w1, cs);
    float p11 = prob_cls(pred, nbase, h1 * WWI + w1, cs);

    float a0 = p00 * (1.f - wh) + p10 * wh;        // zoom H first
    float a1 = p01 * (1.f - wh) + p11 * wh;
    float v  = a0 * (1.f - ww) + a1 * ww;          // then zoom W

    P[t] = (tg != 255) ? v : 2.0f;                 // invalid sorts last
}

// ---------------------------------------------------------------------------
// Kernel 2: exact k-th order statistic by binary search on IEEE bit pattern
// (all values >= 0 so uint compare is monotone). Single block, LDS reduce.
// ---------------------------------------------------------------------------
__device__ int count_le(const float* __restrict__ P, unsigned pat,
                        int tid, int* red) {
    int c = 0;
    for (int k = tid; k < DSN; k += 256)
        c += (__float_as_uint(P[k]) <= pat) ? 1 : 0;
    red[tid] = c;
    __syncthreads();
#pragma unroll
    for (int s = 128; s > 0; s >>= 1) {
        if (tid < s) red[tid] += red[tid + s];
        __syncthreads();
    }
    int r = red[0];
    __syncthreads();
    return r;
}

__global__ void ohem_select(const float* __restrict__ P,
                            float* __restrict__ thrOut) {
    __shared__ int red[256];
    int tid = threadIdx.x;

    // num_valid: valid probs are <= ~1.0, invalid sentinel is 2.0
    int nv = count_le(P, 0x3FC00000u /*1.5f*/, tid, red);

    int k    = max(min(nv, MIN_KEPT_DS) - 1, 0);
    int rank = k + 1;                               // rank-th smallest

    unsigned lo = 0u, hi = 0x40000000u;             // bits(2.0f); cntLE(hi)=DSN>=rank
    for (int it = 0; it < 31; ++it) {
        unsigned mid = lo + ((hi - lo) >> 1);
        int c = count_le(P, mid, tid, red);
        if (c >= rank) hi = mid; else lo = mid + 1;
    }
    float kth = __uint_as_float(lo);
    float thr = (MIN_KEPT_DS >= nv) ? 1.0f : (kth > THR07 ? kth : THR07);
    if (tid == 0) thrOut[0] = thr;
}

// ---------------------------------------------------------------------------
// Kernel 3: hot pass. 19 coalesced channel loads per pixel (128B/wave/chan),
// logsumexp, kept-mask, WMMA wave reduction, per-block partials.
// 4096 blocks x 256 threads x 4 pixels = 4,194,304 (exact, EXEC always full).
// ---------------------------------------------------------------------------
__global__ void ohem_loss(const float* __restrict__ pred,
                          const int* __restrict__ tgt,
                          const float* __restrict__ thrP,
                          float* __restrict__ psum,
                          float* __restrict__ pcnt) {
    const float thr = thrP[0];
    const int tid = threadIdx.x;
    const int stride = gridDim.x * blockDim.x;      // 1,048,576
    int g = blockIdx.x * blockDim.x + tid;

    float accS = 0.f, accC = 0.f;
#pragma unroll 1
    for (int it = 0; it < 4; ++it, g += stride) {
        int n   = g >> 19;                          // / HW
        int rem = g & (HW - 1);
        const float* base = pred + n * (CC * HW) + rem;
        int tg = tgt[g];
        int cs = min(max(tg, 0), CC - 1);

        float xs[CC];
        float m = -3.402823466e38f;
#pragma unroll
        for (int c = 0; c < CC; ++c) {
            xs[c] = base[c * HW];
            m = fmaxf(m, xs[c]);
        }
        float s = 0.f, xt = xs[0];
#pragma unroll
        for (int c = 0; c < CC; ++c) {
            s += expf(xs[c] - m);
            if (c == cs) xt = xs[c];
        }
        float lse = m + logf(s);
        float nll = lse - xt;                       // -log_softmax[true]
        float pf  = expf(xt - m) / s;               // softmax prob of true class
        bool kept = (tg != 255) && (pf <= thr);
        accS += kept ? nll : 0.f;
        accC += kept ? 1.f : 0.f;
    }

    float wS = wave_sum32(accS);                    // matrix-pipe reduction
    float wC = wave_sum32(accC);

    __shared__ float sS[8], sC[8];
    int wid = tid >> 5;
    if ((tid & 31) == 0) { sS[wid] = wS; sC[wid] = wC; }
    __syncthreads();
    if (tid == 0) {
        float tS = 0.f, tC = 0.f;
#pragma unroll
        for (int w = 0; w < 8; ++w) { tS += sS[w]; tC += sC[w]; }
        psum[blockIdx.x] = tS;
        pcnt[blockIdx.x] = tC;
    }
}

// ---------------------------------------------------------------------------
// Kernel 4: single-wave final reduce (fixed order => deterministic) + divide.
// ---------------------------------------------------------------------------
__global__ void ohem_final(const float* __restrict__ psum,
                           const float* __restrict__ pcnt,
                           float* __restrict__ out) {
    int lane = threadIdx.x;
    float aS = 0.f, aC = 0.f;
    for (int i = lane; i < 4096; i += 32) { aS += psum[i]; aC += pcnt[i]; }
    float tS = wave_sum32(aS);
    float tC = wave_sum32(aC);
    if (lane == 0) out[0] = tS / fmaxf(tC, 1.0f);
}

// ---------------------------------------------------------------------------
// ws layout (bytes):
//   [0,        262144)  P: 65536 f32 downsampled true-class probs
//   [262144,   262148)  threshold (1 f32)
//   [262400,   278784)  psum: 4096 f32 block partial nll sums
//   [278784,   295168)  pcnt: 4096 f32 block partial kept counts
// ---------------------------------------------------------------------------
extern "C" void kernel_launch(void* const* d_in, const int* in_sizes, int n_in,
                              void* d_out, int out_size, void* d_ws, size_t ws_size,
                              hipStream_t stream) {
    const float* predict = (const float*)d_in[0];   // (8,19,512,1024) f32
    const int*   target  = (const int*)d_in[1];     // (8,512,1024)    i32
    float*       out     = (float*)d_out;           // scalar f32 loss

    char*  ws   = (char*)d_ws;
    float* P    = (float*)(ws + 0);
    float* thr  = (float*)(ws + 262144);
    float* psum = (float*)(ws + 262400);
    float* pcnt = (float*)(ws + 278784);

    ohem_ds_pred<<<DSN / 256, 256, 0, stream>>>(predict, target, P);
    ohem_select <<<1,         256, 0, stream>>>(P, thr);
    ohem_loss   <<<4096,      256, 0, stream>>>(predict, target, thr, psum, pcnt);
    ohem_final  <<<1,          32, 0, stream>>>(psum, pcnt, out);
}